// mixed_decayRNN_fixin_72464688218245
// MI455X (gfx1250) — compile-verified
//
#include <hip/hip_runtime.h>
#include <hip/hip_bf16.h>

typedef __attribute__((ext_vector_type(2))) float v2f;
typedef __attribute__((ext_vector_type(8))) float v8f;

#define RNN_B 128
#define RNN_T 800
#define RNN_I 9
#define RNN_H 1024
#define RNN_O 8
#define RNN_ALPHA 0.1f

// WGP-scope prefetch (SCOPE 0: pull into all cache levels, stop on WGP hit).
// The __builtin_prefetch lowering picks SCOPE_SE (GL2-only), which is useless
// for data that is already L2-resident — inline asm gives us the WGP variant.
__device__ __forceinline__ void prefetch_wgp(const void* p) {
    asm volatile("global_prefetch_b8 %0, off" :: "v"(p));
}

// ---------------------------------------------------------------------------
// Pack Wrec [H,H] row-major into WMMA-B-operand order:
//   Wpack[(ct*256 + kk)*64 + lane*2 + j] = Wrec[ct*16 + (lane&15)][4*kk + 2*(lane>>4) + j]
// so each wave's B fetch for one V_WMMA_F32_16X16X4_F32 is a single coalesced
// 256-byte block (global_load_b64 per lane). 4 MB, stays resident in L2.
// ---------------------------------------------------------------------------
__global__ __launch_bounds__(256) void pack_wrec_kernel(const float* __restrict__ Wrec,
                                                        float* __restrict__ Wpack) {
    int i    = blockIdx.x * 256 + threadIdx.x;     // 0 .. H*H-1
    int ct   = i >> 14;                            // column tile (64 tiles of 16)
    int rem  = i & 16383;
    int kk   = rem >> 6;                           // k-step (256 steps of 4)
    int lp   = rem & 63;
    int lane = lp >> 1;
    int j    = lp & 1;
    int n    = lane & 15;
    int koff = (lane >> 4) << 1;                   // lanes 16-31 hold K+2,K+3
    Wpack[i] = Wrec[(ct * 16 + n) * RNN_H + kk * 4 + koff + j];
}

// ---------------------------------------------------------------------------
// drive[b,t,h] = sum_i x[b,t,i]*Win[h,i] + 0.1*noise[b,t,h]
// Written into the hiddens slot of d_out; rnn_scan overwrites it in place.
// Pure bandwidth (~840 MB); non-temporal so it doesn't evict hot L2 lines.
// ---------------------------------------------------------------------------
__global__ __launch_bounds__(256) void drive_kernel(const float* __restrict__ x,
                                                    const float* __restrict__ noise,
                                                    const float* __restrict__ Win,
                                                    float* __restrict__ drive) {
    long gid  = (long)blockIdx.x * 256 + threadIdx.x;   // over B*T*H
    int  hcol = (int)(gid & (RNN_H - 1));
    long bt   = gid >> 10;
    const float* xr = x + bt * RNN_I;
    const float* wr = Win + hcol * RNN_I;
    float acc = 0.f;
#pragma unroll
    for (int i = 0; i < RNN_I; ++i) acc += xr[i] * wr[i];
    float nz = __builtin_nontemporal_load(&noise[gid]);
    __builtin_nontemporal_store(acc + 0.1f * nz, &drive[gid]);
}

// ---------------------------------------------------------------------------
// Sequential scan. One workgroup per 16-row batch tile (8 WGs total).
// 32 waves, each owning two 16x16 output tiles of H. h_{t-1} lives in LDS
// (64KB, XOR-swizzled: phys_col = col ^ (4*row) -> conflict-free ds_load_b64
// for the A operand). Each wave keeps its own h_old tile in registers.
// Per step per wave: 256 k-steps x 2 tiles of v_wmma_f32_16x16x4_f32.
// B stream: one WGP-scope prefetch per 1KB block per stream, 8KB ahead.
// hiddens writes / drive reads are non-temporal (streaming, >L2).
// ---------------------------------------------------------------------------
__global__ __launch_bounds__(1024) void rnn_scan_kernel(const float* __restrict__ Wpack,
                                                        float* __restrict__ hid) {
    __shared__ float hbuf[16 * RNN_H];            // 64 KB

    const int bt0  = blockIdx.x * 16;
    const int lane = threadIdx.x & 31;
    const int wave = threadIdx.x >> 5;
    const int n    = lane & 15;                   // N column within tile / A row
    const int hi   = lane >> 4;                   // half-wave select
    const int koff = hi * 2;                      // A/B: lanes 16-31 carry K+2,K+3
    const int c0   = (wave * 2) * 16;             // first owned column tile
    const int c1   = c0 + 16;                     // second owned column tile

    for (int i = threadIdx.x; i < 16 * RNN_H; i += 1024) hbuf[i] = 0.f;
    __syncthreads();

    v8f acc0, acc1, h0, h1;
#pragma unroll
    for (int r = 0; r < 8; ++r) { h0[r] = 0.f; h1[r] = 0.f; }

    const float* wp0 = Wpack + (long)(c0 >> 4) * (256 * 64);
    const float* wp1 = Wpack + (long)(c1 >> 4) * (256 * 64);

    for (int t = 0; t < RNN_T; ++t) {
        // C init = drive tile (still untouched at position t); streaming read
#pragma unroll
        for (int r = 0; r < 8; ++r) {
            long rowoff = ((long)(bt0 + hi * 8 + r) * RNN_T + t) * RNN_H;
            acc0[r] = __builtin_nontemporal_load(&hid[rowoff + c0 + n]);
            acc1[r] = __builtin_nontemporal_load(&hid[rowoff + c1 + n]);
        }

        // pre += h_{t-1} @ Wrec^T over K=1024: 64 blocks x 4 k-steps
        for (int kb = 0; kb < RNN_H / 16; ++kb) {
            // 32 lanes x 32B stride = full 1KB block (4 k-steps), 8 blocks ahead
            const int pkb = ((kb + 8) & 63) * 256 + lane * 8;
            prefetch_wgp(&wp0[pkb]);
            prefetch_wgp(&wp1[pkb]);
#pragma unroll
            for (int u = 0; u < 4; ++u) {
                const int kk = kb * 4 + u;
                const int k  = kk * 4 + koff;
                const v2f a  = *(const v2f*)&hbuf[n * RNN_H + (k ^ (n * 4))];
                const v2f b0 = *(const v2f*)&wp0[kk * 64 + lane * 2];
                const v2f b1 = *(const v2f*)&wp1[kk * 64 + lane * 2];
                acc0 = __builtin_amdgcn_wmma_f32_16x16x4_f32(false, a, false, b0,
                                                             (short)0, acc0, false, false);
                acc1 = __builtin_amdgcn_wmma_f32_16x16x4_f32(false, a, false, b1,
                                                             (short)0, acc1, false, false);
            }
        }

        __syncthreads();   // everyone done reading h_{t-1} from LDS

#pragma unroll
        for (int r = 0; r < 8; ++r) {
            const int m = hi * 8 + r;             // C/D row for this VGPR/half
            long rowoff = ((long)(bt0 + m) * RNN_T + t) * RNN_H;
            float hn0 = fmaxf(acc0[r], 0.f) * RNN_ALPHA + (1.f - RNN_ALPHA) * h0[r];
            float hn1 = fmaxf(acc1[r], 0.f) * RNN_ALPHA + (1.f - RNN_ALPHA) * h1[r];
            h0[r] = hn0;
            h1[r] = hn1;
            hbuf[m * RNN_H + ((c0 + n) ^ (m * 4))] = hn0;
            hbuf[m * RNN_H + ((c1 + n) ^ (m * 4))] = hn1;
            __builtin_nontemporal_store(hn0, &hid[rowoff + c0 + n]);  // persist hiddens
            __builtin_nontemporal_store(hn1, &hid[rowoff + c1 + n]);
        }
        __syncthreads();   // h_t visible before next step's A reads
    }
}

// ---------------------------------------------------------------------------
// logits = hiddens @ Wout^T (O=8), softmax over O. One wave per (b,t) row,
// per-lane partial dot + xor-shuffle reduction (wave32). hiddens read is
// single-use streaming -> non-temporal.
// ---------------------------------------------------------------------------
__global__ __launch_bounds__(256) void out_kernel(const float* __restrict__ hid,
                                                  const float* __restrict__ Wout,
                                                  float* __restrict__ outs) {
    const int wave = threadIdx.x >> 5;
    const int lane = threadIdx.x & 31;
    const long bt  = (long)blockIdx.x * 8 + wave;
    const float* hrow = hid + bt * RNN_H;

    float acc[RNN_O];
#pragma unroll
    for (int o = 0; o < RNN_O; ++o) acc[o] = 0.f;

    for (int j = 0; j < RNN_H / 32; ++j) {
        const int   idx = lane + 32 * j;
        const float hv  = __builtin_nontemporal_load(&hrow[idx]);
#pragma unroll
        for (int o = 0; o < RNN_O; ++o) acc[o] += hv * Wout[o * RNN_H + idx];
    }
#pragma unroll
    for (int o = 0; o < RNN_O; ++o)
        for (int s = 16; s > 0; s >>= 1) acc[o] += __shfl_xor(acc[o], s, 32);

    if (lane == 0) {
        float mx = acc[0];
#pragma unroll
        for (int o = 1; o < RNN_O; ++o) mx = fmaxf(mx, acc[o]);
        float e[RNN_O], sum = 0.f;
#pragma unroll
        for (int o = 0; o < RNN_O; ++o) { e[o] = __expf(acc[o] - mx); sum += e[o]; }
        const float inv = 1.f / sum;
#pragma unroll
        for (int o = 0; o < RNN_O; ++o) outs[bt * RNN_O + o] = e[o] * inv;
    }
}

// ---------------------------------------------------------------------------
extern "C" void kernel_launch(void* const* d_in, const int* in_sizes, int n_in,
                              void* d_out, int out_size, void* d_ws, size_t ws_size,
                              hipStream_t stream) {
    const float* x     = (const float*)d_in[0];   // [B,T,I]
    const float* noise = (const float*)d_in[1];   // [B,T,H]
    const float* Win   = (const float*)d_in[2];   // [H,I]
    const float* Wrec  = (const float*)d_in[3];   // [H,H]
    const float* Wout  = (const float*)d_in[4];   // [O,H]

    float* hid  = (float*)d_out;                            // [B,T,H] hiddens
    float* outs = hid + (long)RNN_B * RNN_T * RNN_H;        // [B,T,O] softmax
    float* Wpack = (float*)d_ws;                            // 4 MB packed Wrec

    pack_wrec_kernel<<<(RNN_H * RNN_H) / 256, 256, 0, stream>>>(Wrec, Wpack);
    drive_kernel<<<(int)(((long)RNN_B * RNN_T * RNN_H) / 256), 256, 0, stream>>>(x, noise, Win, hid);
    rnn_scan_kernel<<<RNN_B / 16, 1024, 0, stream>>>(Wpack, hid);
    out_kernel<<<(RNN_B * RNN_T) / 8, 256, 0, stream>>>(hid, Wout, outs);
}